// DetectionLoss_54494545052107
// MI455X (gfx1250) — compile-verified
//
#include <hip/hip_runtime.h>
#include <hip/hip_bf16.h>
#include <math.h>

typedef __attribute__((ext_vector_type(2))) float v2f;
typedef __attribute__((ext_vector_type(8))) float v8f;

#define IOU_TH 0.5f
#define EPSF   1e-6f

__device__ __forceinline__ float iou_box(float4 p, float4 q) {
    float ltx = fmaxf(p.x, q.x), lty = fmaxf(p.y, q.y);
    float rbx = fminf(p.z, q.z), rby = fminf(p.w, q.w);
    float w = fmaxf(rbx - ltx, 0.f), h = fmaxf(rby - lty, 0.f);
    float inter = w * h;
    float a1 = (p.z - p.x) * (p.w - p.y);
    float a2 = (q.z - q.x) * (q.w - q.y);
    return inter / (a1 + a2 - inter + EPSF);
}

__device__ __forceinline__ float smooth_l1(float x) {
    float ax = fabsf(x);
    return (ax < 1.f) ? 0.5f * x * x : ax - 0.5f;
}

// ---------------- Kernel 1: per-anchor best GT (max + first-argmax over G) ---------
__global__ void k_match(const float4* __restrict__ anchors, const float4* __restrict__ gt,
                        float* __restrict__ bgiou, int* __restrict__ bgidx, int A, int G) {
    int b = blockIdx.y;
    int a = blockIdx.x * 256 + threadIdx.x;
    __shared__ float4 gts[64];
    if (threadIdx.x < G) gts[threadIdx.x] = gt[b * G + threadIdx.x];
    __syncthreads();
    __builtin_prefetch(&anchors[a + 256], 0, 0);
    float4 ab = anchors[a];
    float best = -1.f; int bi = 0;
    for (int g = 0; g < G; ++g) {
        float v = iou_box(ab, gts[g]);
        if (v > best) { best = v; bi = g; }   // strict > keeps first index (jnp.argmax)
    }
    bgiou[(long)b * A + a] = best;
    bgidx[(long)b * A + a] = bi;
}

// ---------------- Kernel 2: per-GT best anchor (first-argmax over A) ---------------
__global__ void k_best_anchor(const float4* __restrict__ anchors, const float4* __restrict__ gt,
                              int* __restrict__ bai, int A, int G) {
    int b = blockIdx.y, g = blockIdx.x;
    float4 gb = gt[b * G + g];
    unsigned long long best = 0ULL;
    for (int a = threadIdx.x; a < A; a += 256) {
        __builtin_prefetch(&anchors[a + 256], 0, 0);
        float v = iou_box(anchors[a], gb);                  // v >= 0 -> bits are order-preserving
        unsigned long long key = ((unsigned long long)__float_as_uint(v) << 32) |
                                 (unsigned long long)(0xFFFFFFFFu - (unsigned)a); // smaller a wins ties
        if (key > best) best = key;
    }
    __shared__ unsigned long long red[256];
    red[threadIdx.x] = best;
    __syncthreads();
    for (int s = 128; s > 0; s >>= 1) {
        if (threadIdx.x < s)
            red[threadIdx.x] = (red[threadIdx.x] >= red[threadIdx.x + s]) ? red[threadIdx.x]
                                                                          : red[threadIdx.x + s];
        __syncthreads();
    }
    if (threadIdx.x == 0)
        bai[b * G + g] = (int)(0xFFFFFFFFu - (unsigned)(red[0] & 0xFFFFFFFFu));
}

// ---------------- Kernel 3: per-anchor losses + block partial sums -----------------
__global__ void k_loss(const float4* __restrict__ bbox, const float* __restrict__ conf,
                       const float4* __restrict__ gt, const float* __restrict__ bgiou,
                       const int* __restrict__ bgidx, const int* __restrict__ bai,
                       float* __restrict__ neg, float4* __restrict__ partials,
                       int A, int G) {
    int b = blockIdx.y;
    int tid = threadIdx.x;
    int a = blockIdx.x * 256 + tid;
    __shared__ float4 gts[64];
    __shared__ int bsh[64];
    if (tid < G) { gts[tid] = gt[b * G + tid]; bsh[tid] = bai[b * G + tid]; }
    __syncthreads();

    long idx = (long)b * A + a;
    __builtin_prefetch(&conf[idx], 0, 0);
    float bg = bgiou[idx];
    int gi = bgidx[idx];
    bool pos = bg > IOU_TH;
    for (int g = 0; g < G; ++g) pos = pos || (bsh[g] == a);   // .at[best_anchor_idx].set(True)

    float4 bp = bbox[idx];
    float4 m  = gts[gi];
    // center/size smooth-L1
    float pcx = (bp.x + bp.z) * 0.5f, pcy = (bp.y + bp.w) * 0.5f;
    float psx = bp.z - bp.x,          psy = bp.w - bp.y;
    float gcx = (m.x + m.z) * 0.5f,   gcy = (m.y + m.w) * 0.5f;
    float gsx = m.z - m.x,            gsy = m.w - m.y;
    float center_l = smooth_l1(pcx - gcx) + smooth_l1(pcy - gcy);
    float size_l   = smooth_l1(psx - gsx) + smooth_l1(psy - gsy);
    // GIoU (reference's nonstandard union)
    float iou = iou_box(bp, m);
    float eltx = fminf(bp.x, m.x), elty = fminf(bp.y, m.y);
    float erbx = fmaxf(bp.z, m.z), erby = fmaxf(bp.w, m.w);
    float ew = fmaxf(erbx - eltx, 0.f), eh = fmaxf(erby - elty, 0.f);
    float enc = ew * eh + EPSF;
    float pa = psx * psy, ta = gsx * gsy;
    float uni = pa + ta - iou * pa * ta;
    float giou = iou - (enc - uni) / enc;
    float box_loss = 0.5f * (center_l + size_l) + 0.5f * (1.f - giou);
    // focal
    float cp = conf[idx];
    float t = pos ? iou : 0.f;
    float p = fminf(fmaxf(cp, 1e-7f), 1.f - 1e-7f);
    float bce = -(t * logf(p) + (1.f - t) * log1pf(-p));
    float pt = (t > 0.f) ? cp : 1.f - cp;                    // unclipped pred, per reference
    float omp = 1.f - pt;
    float wgt = omp * omp * ((t > 0.f) ? 0.25f : 0.75f);
    float f = wgt * bce;

    float posf = pos ? 1.f : 0.f;
    neg[idx] = pos ? 0.f : f;

    float4 val;
    val.x = box_loss * posf;  // loc
    val.y = f * posf;         // positive focal
    val.z = iou * posf;       // sum iou of positives
    val.w = posf;             // npos (exact small-int floats)
    __shared__ float4 red[256];
    red[tid] = val;
    __syncthreads();
    for (int s = 128; s > 0; s >>= 1) {
        if (tid < s) {
            red[tid].x += red[tid + s].x; red[tid].y += red[tid + s].y;
            red[tid].z += red[tid + s].z; red[tid].w += red[tid + s].w;
        }
        __syncthreads();
    }
    if (tid == 0) partials[(long)b * gridDim.x + blockIdx.x] = red[0];
}

// ---------------- Kernel 4: per-sample deterministic reduction of partials ---------
__global__ void k_reduce(const float4* __restrict__ partials, float* __restrict__ sums, int nchunk) {
    int b = blockIdx.x, tid = threadIdx.x;
    __shared__ float4 red[128];
    float4 v = make_float4(0.f, 0.f, 0.f, 0.f);
    for (int i = tid; i < nchunk; i += 128) {
        float4 pv = partials[(long)b * nchunk + i];
        v.x += pv.x; v.y += pv.y; v.z += pv.z; v.w += pv.w;
    }
    red[tid] = v;
    __syncthreads();
    for (int s = 64; s > 0; s >>= 1) {
        if (tid < s) {
            red[tid].x += red[tid + s].x; red[tid].y += red[tid + s].y;
            red[tid].z += red[tid + s].z; red[tid].w += red[tid + s].w;
        }
        __syncthreads();
    }
    if (tid == 0) {
        sums[b * 8 + 0] = red[0].x;  // loc
        sums[b * 8 + 1] = red[0].y;  // fpos
        sums[b * 8 + 2] = red[0].z;  // siou
        sums[b * 8 + 3] = red[0].w;  // nposf
    }
}

// ---------------- Kernel 5: exact top-k sum via 4-pass MSD radix select ------------
__global__ void k_select(const float* __restrict__ neg, float* __restrict__ sums, int A) {
    int b = blockIdx.x, tid = threadIdx.x;
    __shared__ unsigned int hist[256];
    __shared__ float fred[256];
    __shared__ unsigned int s_t, s_c;
    const float* nb = neg + (long)b * A;

    int npos = (int)(sums[b * 8 + 3] + 0.5f);
    int nneg = min(npos * 3, A - npos);
    float result = 0.f;
    if (nneg > 0) {                      // uniform across block
        unsigned int prefix = 0, k = (unsigned)nneg;
        for (int pass = 0; pass < 4; ++pass) {
            int shift = 24 - 8 * pass;
            unsigned int mask_hi = (pass == 0) ? 0u : (0xFFFFFFFFu << (shift + 8));
            hist[tid] = 0;
            __syncthreads();
            for (int a = tid; a < A; a += 256) {
                unsigned int v = __float_as_uint(nb[a]);   // all values >= +0.0
                if ((v & mask_hi) == prefix) atomicAdd(&hist[(v >> shift) & 255u], 1u);
            }
            __syncthreads();
            if (tid == 0) {
                unsigned int c = 0; int t = 0;
                for (int j = 255; j >= 0; --j) {
                    unsigned int h = hist[j];
                    if (c + h >= k) { t = j; break; }      // Cgt(t) < k <= Cgt(t)+hist[t]
                    c += h;
                }
                s_t = (unsigned)t; s_c = c;
            }
            __syncthreads();
            prefix |= (s_t << shift);
            k -= s_c;
            __syncthreads();
        }
        unsigned int T = prefix;          // exact bits of k-th largest value
        float local = 0.f;
        for (int a = tid; a < A; a += 256) {
            float v = nb[a];
            if (__float_as_uint(v) > T) local += v;
        }
        fred[tid] = local;
        __syncthreads();
        for (int s = 128; s > 0; s >>= 1) {
            if (tid < s) fred[tid] += fred[tid + s];
            __syncthreads();
        }
        result = fred[0] + (float)k * __uint_as_float(T);   // k copies of the tie value
    }
    if (tid == 0) sums[b * 8 + 4] = result;
}

// ---------------- Kernel 6: finalize via WMMA (one wave, EXEC all-1s) --------------
// D = A(16x4) x ones(4x16) + C, chained 4x: row m of D = total of quantity m over
// 16 samples. A-layout per ISA: lane l holds M=l%16, K = (l<16 ? {0,1} : {2,3}).
// B = all-ones is layout-invariant. D read from documented C/D layout.
__global__ void k_finalize(const float* __restrict__ sums, float* __restrict__ out, int Bn) {
    int l = threadIdx.x;
    int m = l & 15;
    int kb = (l < 16) ? 0 : 2;
    v8f c = {0.f, 0.f, 0.f, 0.f, 0.f, 0.f, 0.f, 0.f};
    v2f ones; ones.x = 1.f; ones.y = 1.f;
    for (int i = 0; i < 4; ++i) {
        int s0 = 4 * i + kb;
        v2f av;
        av.x = (m < 5 && s0     < Bn) ? sums[s0 * 8 + m]       : 0.f;
        av.y = (m < 5 && s0 + 1 < Bn) ? sums[(s0 + 1) * 8 + m] : 0.f;
        c = __builtin_amdgcn_wmma_f32_16x16x4_f32(false, av, false, ones,
                                                  (short)0, c, false, false);
    }
    if (l == 0) {
        float loc = c[0], fpos = c[1], siou = c[2], nposf = c[3], negs = c[4];
        float denom = fmaxf(1.f, nposf);
        float tl = loc / denom;
        float tc = (negs + fpos) / denom;
        float mi = siou / denom;
        out[0] = tl + tc;   // total_loss
        out[1] = tc;        // total_conf
        out[2] = tl;        // total_loc
        out[3] = mi;        // mean_iou
    }
}

extern "C" void kernel_launch(void* const* d_in, const int* in_sizes, int n_in,
                              void* d_out, int out_size, void* d_ws, size_t ws_size,
                              hipStream_t stream) {
    const float* bbox    = (const float*)d_in[0];   // [B,A,4]
    const float* conf    = (const float*)d_in[1];   // [B,A]
    const float* anchors = (const float*)d_in[2];   // [A,4]
    const float* gt      = (const float*)d_in[3];   // [B,G,4]

    int A  = in_sizes[2] / 4;
    int Bn = in_sizes[1] / A;
    int G  = in_sizes[3] / (Bn * 4);
    int nchunk = A / 256;

    float* ws = (float*)d_ws;
    size_t BA = (size_t)Bn * A;
    float*  bgiou    = ws;
    int*    bgidx    = (int*)(ws + BA);
    float*  negbuf   = ws + 2 * BA;
    int*    bai      = (int*)(ws + 3 * BA);
    float4* partials = (float4*)(ws + 3 * BA + (size_t)Bn * G);
    float*  sums     = ws + 3 * BA + (size_t)Bn * G + (size_t)Bn * nchunk * 4;

    dim3 gAnch(nchunk, Bn);
    k_match<<<gAnch, 256, 0, stream>>>((const float4*)anchors, (const float4*)gt,
                                       bgiou, bgidx, A, G);
    dim3 gGT(G, Bn);
    k_best_anchor<<<gGT, 256, 0, stream>>>((const float4*)anchors, (const float4*)gt,
                                           bai, A, G);
    k_loss<<<gAnch, 256, 0, stream>>>((const float4*)bbox, conf, (const float4*)gt,
                                      bgiou, bgidx, bai, negbuf, partials, A, G);
    k_reduce<<<Bn, 128, 0, stream>>>(partials, sums, nchunk);
    k_select<<<Bn, 256, 0, stream>>>(negbuf, sums, A);
    k_finalize<<<1, 32, 0, stream>>>(sums, (float*)d_out, Bn);
}